// AntiBiasL1Loss_6700148982004
// MI455X (gfx1250) — compile-verified
//
#include <hip/hip_runtime.h>

typedef __attribute__((ext_vector_type(2))) float v2f;
typedef __attribute__((ext_vector_type(8))) float v8f;

#define NUM_GRADES 5
#define BLOCKS 1024
#define THREADS 256

// ---------------------------------------------------------------------------
// Kernel 1: streaming pass. float4 (global_load_b128) loads, per-thread
// 5 sums + 5 counts in VGPRs, wave shfl_xor tree, LDS combine across the
// 8 waves of the block, one 16-float partial row per block into d_ws.
// ---------------------------------------------------------------------------
__device__ __forceinline__ void accum_elem(float p, float t,
                                           float (&sums)[NUM_GRADES],
                                           float (&cnts)[NUM_GRADES]) {
    float err = fabsf(p - t);
#pragma unroll
    for (int g = 0; g < NUM_GRADES; ++g) {
        // y_true is an exactly-representable integer grade; compare directly.
        float m = (t == (float)g) ? 1.0f : 0.0f;   // v_cmp + v_cndmask
        sums[g] = fmaf(m, err, sums[g]);           // v_fmac
        cnts[g] += m;                              // v_add
    }
}

__global__ __launch_bounds__(THREADS)
void abl1_partials(const float* __restrict__ yp, const float* __restrict__ yt,
                   float* __restrict__ P, int N) {
    float sums[NUM_GRADES] = {0.f, 0.f, 0.f, 0.f, 0.f};
    float cnts[NUM_GRADES] = {0.f, 0.f, 0.f, 0.f, 0.f};

    const int tid = blockIdx.x * blockDim.x + threadIdx.x;
    const int T   = gridDim.x * blockDim.x;
    const int n4  = N >> 2;

    const float4* p4 = (const float4*)yp;
    const float4* t4 = (const float4*)yt;

    for (int i = tid; i < n4; i += T) {
        float4 p = p4[i];
        float4 t = t4[i];
        accum_elem(p.x, t.x, sums, cnts);
        accum_elem(p.y, t.y, sums, cnts);
        accum_elem(p.z, t.z, sums, cnts);
        accum_elem(p.w, t.w, sums, cnts);
    }
    // scalar tail (N % 4)
    for (int i = (n4 << 2) + tid; i < N; i += T)
        accum_elem(yp[i], yt[i], sums, cnts);

    // wave32 butterfly reduction
#pragma unroll
    for (int off = 16; off > 0; off >>= 1) {
#pragma unroll
        for (int g = 0; g < NUM_GRADES; ++g) {
            sums[g] += __shfl_xor(sums[g], off, 32);
            cnts[g] += __shfl_xor(cnts[g], off, 32);
        }
    }

    __shared__ float red[THREADS / 32][16];
    const int lane = threadIdx.x & 31;
    const int wave = threadIdx.x >> 5;
    if (lane == 0) {
#pragma unroll
        for (int g = 0; g < NUM_GRADES; ++g) {
            red[wave][g]              = sums[g];
            red[wave][NUM_GRADES + g] = cnts[g];
        }
#pragma unroll
        for (int k = 2 * NUM_GRADES; k < 16; ++k) red[wave][k] = 0.0f;
    }
    __syncthreads();

    if (threadIdx.x < 16) {
        float a = 0.0f;
#pragma unroll
        for (int w = 0; w < THREADS / 32; ++w) a += red[w][threadIdx.x];
        P[blockIdx.x * 16 + threadIdx.x] = a;   // cols 0-4 sums, 5-9 counts, rest 0
    }
}

// ---------------------------------------------------------------------------
// Kernel 2: one wave. Each lane sums 512 partial rows for its column
// (independent, coalesced loads), then one v_wmma_f32_16x16x4_f32 with a
// ones A-matrix folds the two half-wave partials per column (full-f32,
// deterministic). Lane 0 computes the present-bin mean-of-means.
// ---------------------------------------------------------------------------
__global__ __launch_bounds__(32)
void abl1_final(const float* __restrict__ P, float* __restrict__ out, int rows) {
    const int lane = threadIdx.x;      // 0..31, all active -> EXEC all-ones
    const int col  = lane & 15;
    const int half = lane >> 4;        // 0 or 1

    float ps = 0.0f;
#pragma unroll 8
    for (int r = half; r < rows; r += 2)
        ps += P[r * 16 + col];

    // B (4x16 f32): lanes 0-15 hold K=0 (b.x) / K=1 (b.y); lanes 16-31 hold
    // K=2 / K=3. Put each half-wave's partial in b.x, zero in b.y, so the
    // K-reduction D[m][n] = sum_k B[k][n] yields the full column sum.
    v2f a; a.x = 1.0f; a.y = 1.0f;     // A = ones(16x4)
    v2f b; b.x = ps;   b.y = 0.0f;
    v8f c = {};
    c = __builtin_amdgcn_wmma_f32_16x16x4_f32(
            /*neg_a=*/false, a, /*neg_b=*/false, b,
            /*c_mod=*/(short)0, c, /*reuse_a=*/false, /*reuse_b=*/false);

    // C/D layout: lane L<16, VGPR0 = D[0][L] = colsum[L] (rows identical).
    float cs = c[0];

    float sums[NUM_GRADES], cnts[NUM_GRADES];
#pragma unroll
    for (int g = 0; g < NUM_GRADES; ++g) {
        sums[g] = __shfl(cs, g, 32);
        cnts[g] = __shfl(cs, NUM_GRADES + g, 32);
    }

    if (lane == 0) {
        float acc = 0.0f, np = 0.0f;
#pragma unroll
        for (int g = 0; g < NUM_GRADES; ++g) {
            if (cnts[g] > 0.0f) { acc += sums[g] / cnts[g]; np += 1.0f; }
        }
        out[0] = acc / np;
    }
}

// ---------------------------------------------------------------------------
extern "C" void kernel_launch(void* const* d_in, const int* in_sizes, int n_in,
                              void* d_out, int out_size, void* d_ws, size_t ws_size,
                              hipStream_t stream) {
    const float* yp = (const float*)d_in[0];
    const float* yt = (const float*)d_in[1];
    const int N = in_sizes[0];

    float* P = (float*)d_ws;   // BLOCKS x 16 floats = 64 KB scratch

    abl1_partials<<<BLOCKS, THREADS, 0, stream>>>(yp, yt, P, N);
    abl1_final<<<1, 32, 0, stream>>>(P, (float*)d_out, BLOCKS);
}